// NeighborAggregator_1735166787608
// MI455X (gfx1250) — compile-verified
//
#include <hip/hip_runtime.h>
#include <hip/hip_bf16.h>

typedef __attribute__((ext_vector_type(2))) float v2f;
typedef __attribute__((ext_vector_type(4))) float v4f;
typedef __attribute__((ext_vector_type(8))) float v8f;

#define D 128               // D_IN == D_OUT
#define SEG_PER_BLOCK 16    // one 16-row WMMA tile of segments per block
#define THREADS 256         // 8 waves (wave32)
#define LDS_STRIDE 132      // 128 + 4 pad -> rows land on distinct banks

__global__ __launch_bounds__(THREADS) void seg_mean_gemm_fused(
    const int* __restrict__ action,
    const float* __restrict__ feat,        // [total_rows, 128]
    const int* __restrict__ seg_ids,       // [total_rows], sorted ascending
    const float* __restrict__ weight,      // [128, 128] (K-major rows)
    float* __restrict__ out,               // [n_seg, 128]
    int n_seg, int total_rows)
{
    __shared__ float ldsA[SEG_PER_BLOCK * LDS_STRIDE];

    const int tid  = threadIdx.x;
    const int wave = tid >> 5;     // 0..7
    const int lane = tid & 31;
    const int hi16 = lane >> 4;    // 0 for lanes 0-15, 1 for lanes 16-31
    const int l16  = lane & 15;
    const int segBase = blockIdx.x * SEG_PER_BLOCK;

    // ---------------- Phase 1: ragged segment means (bandwidth phase) -------
    // Each wave produces 2 rows of the 16x128 A-tile. One row of 128 f32 is
    // read per wave-iteration as 32 lanes x float4 (coalesced 512B).
    #pragma unroll
    for (int s = 0; s < 2; ++s) {
        const int localRow = wave * 2 + s;        // 0..15
        const int seg = segBase + localRow;
        if (seg < n_seg) {
            // lower_bound: first index with seg_ids[idx] >= seg.
            // Segments are nonempty & contiguous, so this is the row offset.
            int lo = 0, hi = total_rows;
            while (lo < hi) {
                int mid = (lo + hi) >> 1;
                if (seg_ids[mid] < seg) lo = mid + 1; else hi = mid;
            }
            const int start = lo;
            const int size  = action[seg] + 1;    // 1..32 rows

            v4f acc = {0.f, 0.f, 0.f, 0.f};
            const float* rowp = feat + (size_t)start * D + (size_t)lane * 4;
            for (int r = 0; r < size; ++r) {
                acc += *(const v4f*)rowp;         // global_load_b128
                rowp += D;
            }
            const float inv = 1.0f / (float)size;
            acc *= inv;
            *(v4f*)(&ldsA[localRow * LDS_STRIDE + lane * 4]) = acc;
        }
    }
    __syncthreads();

    // ---------------- Phase 2: 16x128 (LDS) x 128x128 (weight) WMMA --------
    // Wave w owns output columns [w*16, w*16+16). K=128 -> 32 x
    // V_WMMA_F32_16X16X4_F32, exact fp32 accumulate.
    const int col0 = wave * 16;

    // Per ISA 32-bit A 16x4 layout: lanes 0-15 hold K={0,1} in v0/v1,
    // lanes 16-31 hold K={2,3}; M = lane&15. Both K values are contiguous
    // in the LDS row -> one ds_load_b64 (8B-aligned: row*132 + k + 2*hi16
    // is even for k%4==0).
    const float* bptr0 = weight + (size_t)(2 * hi16) * D + col0 + l16;

    v8f acc = {};
    #pragma unroll 4
    for (int k = 0; k < D; k += 4) {
        const v2f a = *(const v2f*)(&ldsA[l16 * LDS_STRIDE + k + 2 * hi16]);

        // B 4x16 mirrored layout: v0 holds row K=(k+2*hi16), v1 holds the
        // next K row; N = lane&15. Rows are 512B apart in `weight`.
        v2f b;
        b.x = bptr0[(size_t)k * D];          // W[k + 2*hi16    ][col0+l16]
        b.y = bptr0[(size_t)k * D + D];      // W[k + 2*hi16 + 1][col0+l16]

        acc = __builtin_amdgcn_wmma_f32_16x16x4_f32(
            /*neg_a=*/false, a, /*neg_b=*/false, b,
            /*c_mod=*/(short)0, acc, /*reuse_a=*/false, /*reuse_b=*/false);
    }

    // C/D 16x16 f32 layout: VGPR v -> M = v + 8*hi16, N = l16.
    #pragma unroll
    for (int v = 0; v < 8; ++v) {
        const int m   = v + 8 * hi16;
        const int row = segBase + m;
        if (row < n_seg) {
            out[(size_t)row * D + col0 + l16] = acc[v];
        }
    }
}

extern "C" void kernel_launch(void* const* d_in, const int* in_sizes, int n_in,
                              void* d_out, int out_size, void* d_ws, size_t ws_size,
                              hipStream_t stream) {
    (void)n_in; (void)d_ws; (void)ws_size; (void)out_size;

    const int*   action  = (const int*)d_in[0];
    const float* feat    = (const float*)d_in[1];
    const int*   seg_ids = (const int*)d_in[2];
    const float* weight  = (const float*)d_in[3];
    float*       out     = (float*)d_out;

    const int n_seg      = in_sizes[0];
    const int total_rows = in_sizes[2];      // len(segment_ids)

    const int blocks = (n_seg + SEG_PER_BLOCK - 1) / SEG_PER_BLOCK;
    hipLaunchKernelGGL(seg_mean_gemm_fused, dim3(blocks), dim3(THREADS), 0, stream,
                       action, feat, seg_ids, weight, out, n_seg, total_rows);
}